// LGNCausE_21852793602103
// MI455X (gfx1250) — compile-verified
//
#include <hip/hip_runtime.h>
#include <hip/hip_bf16.h>
#include <math.h>

#define N_USER   100000
#define N_ITEM   50000
#define N_NODES  150000
#define DDIM     64
#define N_LAYERS 3
#define N_EDGES  1200000
#define BATCH    4096
#define LSLOT    5
#define NPAIR    (BATCH * LSLOT)
#define ND       (N_NODES * DDIM)

typedef __attribute__((ext_vector_type(2))) float v2f;
typedef __attribute__((ext_vector_type(8))) float v8f;

// ---------------- utility kernels ----------------

__global__ void k_zero(float* p, int n) {
    int i = blockIdx.x * blockDim.x + threadIdx.x;
    if (i < n) p[i] = 0.0f;
}

__global__ void k_copy(float* dst, const float* src, int n) {
    int i = blockIdx.x * blockDim.x + threadIdx.x;
    if (i < n) dst[i] = src[i];
}

__global__ void k_quarter(float* p, int n) {
    int i = blockIdx.x * blockDim.x + threadIdx.x;
    if (i < n) p[i] *= 0.25f;   // 1/(N_LAYERS+1)
}

// deg[dst[e]] += 1
__global__ void k_deg(float* deg, const int* dst, int nE) {
    int i = blockIdx.x * blockDim.x + threadIdx.x;
    if (i < nE) atomicAdd(&deg[dst[i]], 1.0f);
}

// norm = clip(deg,1,None)^-0.5, in place
__global__ void k_norm(float* d, int n) {
    int i = blockIdx.x * blockDim.x + threadIdx.x;
    if (i < n) {
        float v = d[i];
        v = (v < 1.0f) ? 1.0f : v;
        d[i] = rsqrtf(v);
    }
}

// out[i] = in[i] * norm[node]   (dst may alias src)
__global__ void k_scale(float* out, const float* in, const float* norm, int n) {
    int i = blockIdx.x * blockDim.x + threadIdx.x;
    if (i < n) out[i] = in[i] * norm[i >> 6];
}

// buf[i] *= norm[node]; acc[i] += buf[i]
__global__ void k_scale_add(float* acc, float* buf, const float* norm, int n) {
    int i = blockIdx.x * blockDim.x + threadIdx.x;
    if (i < n) {
        float v = buf[i] * norm[i >> 6];
        buf[i] = v;
        acc[i] += v;
    }
}

// out[dst[e]*64 + 4c .. +3] += hs[src[e]*64 + 4c .. +3]; 16 lanes per edge
__global__ void k_scatter(float* out, const float* hs,
                          const int* src, const int* dst, int nE) {
    int t = blockIdx.x * blockDim.x + threadIdx.x;
    int e = t >> 4;
    if (e < nE) {
        int c = (t & 15) << 2;
        const float4 v = *(const float4*)(hs + (size_t)src[e] * DDIM + c);
        float* dp = out + (size_t)dst[e] * DDIM + c;
        atomicAdd(dp + 0, v.x);
        atomicAdd(dp + 1, v.y);
        atomicAdd(dp + 2, v.z);
        atomicAdd(dp + 3, v.w);
    }
}

// ---------------- WMMA dot-product scoring ----------------
// One wave computes 16 pair scores per feature table using
// V_WMMA_F32_16X16X4_F32 accumulated over 16 K-chunks of 4.
// A (16x4): lanes 0-15 -> M=lane, K0/K1; lanes 16-31 -> M=lane-16, K2/K3.
// B (4x16): VGPR0 holds K=0 (lanes 0-15) / K=2 (lanes 16-31), VGPR1 K+1.
// D (16x16 f32): VGPR v = row M=v (lanes 0-15) / M=v+8 (lanes 16-31);
// diagonal lives in lanes 0-7 (c[lane]) and lanes 24-31 (c[lane-24]).

__device__ __forceinline__ float diag_extract(const v8f& c, int lane) {
    int ll = lane & 7;
    float d = c[0];
    d = (ll == 1) ? c[1] : d;
    d = (ll == 2) ? c[2] : d;
    d = (ll == 3) ? c[3] : d;
    d = (ll == 4) ? c[4] : d;
    d = (ll == 5) ? c[5] : d;
    d = (ll == 6) ? c[6] : d;
    d = (ll == 7) ? c[7] : d;
    return d;
}

__global__ void k_score_wmma(const float* __restrict__ fc,
                             const float* __restrict__ ft,
                             const int* __restrict__ user,
                             const int* __restrict__ item,
                             float* __restrict__ score_c,
                             float* __restrict__ score_t) {
    int tid  = blockIdx.x * blockDim.x + threadIdx.x;
    int wave = tid >> 5;
    int lane = threadIdx.x & 31;
    int pb   = wave * 16;                 // 16 pairs per wave, grid sized exactly
    int m    = lane & 15;                 // A row / B column for this lane
    int koff = (lane < 16) ? 0 : 2;       // which K pair this half-wave feeds
    int p    = pb + m;

    int u  = user[p];
    int it = item[p] + N_USER;
    const float* uc = fc + (size_t)u  * DDIM;
    const float* ic = fc + (size_t)it * DDIM;
    const float* ut = ft + (size_t)u  * DDIM;
    const float* iv = ft + (size_t)it * DDIM;

    v8f cc = {};
    v8f ct = {};
#pragma unroll
    for (int kc = 0; kc < 16; ++kc) {
        int o = 4 * kc + koff;
        v2f a_c = *(const v2f*)(uc + o);
        v2f b_c = *(const v2f*)(ic + o);
        v2f a_t = *(const v2f*)(ut + o);
        v2f b_t = *(const v2f*)(iv + o);
        cc = __builtin_amdgcn_wmma_f32_16x16x4_f32(false, a_c, false, b_c,
                                                   (short)0, cc, false, false);
        ct = __builtin_amdgcn_wmma_f32_16x16x4_f32(false, a_t, false, b_t,
                                                   (short)0, ct, false, false);
    }

    float dc = diag_extract(cc, lane);
    float dt = diag_extract(ct, lane);
    bool valid = (lane < 8) || (lane >= 24);
    int  mo    = (lane < 8) ? lane : (lane - 16);
    if (valid) {
        score_c[pb + mo] = dc;
        score_t[pb + mo] = dt;
    }
}

// ---------------- losses / reductions ----------------
// red layout: 0 bce_c, 1 bce_t, 2 dist_c, 3 dist_t, 4 cnt_ctrl, 5 cnt_trt,
//             6 disc_user, 7 disc_item

__device__ __forceinline__ float bce_logits(float x, float y) {
    return fmaxf(x, 0.0f) - x * y + log1pf(expf(-fabsf(x)));
}

__global__ void k_reduce_loss(const float* __restrict__ sc,
                              const float* __restrict__ st,
                              const float* __restrict__ label,
                              const unsigned char* __restrict__ mask,
                              float* __restrict__ red) {
    int p = blockIdx.x * blockDim.x + threadIdx.x;
    if (p < NPAIR) {
        int   b  = p / LSLOT;
        float y  = label[p];
        float wt = mask[b] ? 1.0f : 0.0f;
        float wc = 1.0f - wt;
        float xc = sc[p], xt = st[p];
        float sgc = 1.0f / (1.0f + expf(-xc));
        float sgt = 1.0f / (1.0f + expf(-xt));
        atomicAdd(&red[0], bce_logits(xc, y) * wc);
        atomicAdd(&red[1], bce_logits(xt, y) * wt);
        atomicAdd(&red[2], fabsf(sgc - y) * wc);
        atomicAdd(&red[3], fabsf(sgt - y) * wt);
        if ((p % LSLOT) == 0) {
            atomicAdd(&red[4], wc);
            atomicAdd(&red[5], wt);
        }
    }
}

__global__ void k_reduce_disc(const float* __restrict__ fc,
                              const float* __restrict__ ft,
                              const int* __restrict__ user,
                              const int* __restrict__ item,
                              float* __restrict__ red) {
    int p = blockIdx.x * blockDim.x + threadIdx.x;
    if (p < NPAIR) {
        size_t uo = (size_t)user[p] * DDIM;
        size_t io = (size_t)(item[p] + N_USER) * DDIM;
        float su = 0.0f, si = 0.0f;
#pragma unroll 4
        for (int d = 0; d < DDIM; d += 4) {
            float4 a = *(const float4*)(fc + uo + d);
            float4 b = *(const float4*)(ft + uo + d);
            float4 c = *(const float4*)(fc + io + d);
            float4 e = *(const float4*)(ft + io + d);
            float dx = a.x - b.x, dy = a.y - b.y, dz = a.z - b.z, dw = a.w - b.w;
            su += dx * dx + dy * dy + dz * dz + dw * dw;
            dx = c.x - e.x; dy = c.y - e.y; dz = c.z - e.z; dw = c.w - e.w;
            si += dx * dx + dy * dy + dz * dz + dw * dw;
        }
        atomicAdd(&red[6], su);
        atomicAdd(&red[7], si);
    }
}

__global__ void k_finalize(const float* __restrict__ red, float* __restrict__ out) {
    if (blockIdx.x == 0 && threadIdx.x == 0) {
        float den_c = fmaxf(red[4] * (float)LSLOT, 1.0f);
        float den_t = fmaxf(red[5] * (float)LSLOT, 1.0f);
        out[0] = red[0] / den_c;                                   // control_loss
        out[1] = red[1] / den_t;                                   // treatment_loss
        out[2] = (red[6] + red[7]) / ((float)NPAIR * (float)DDIM); // discrepancy
        out[3] = red[2] / den_c;                                   // control_distance
        out[4] = red[3] / den_t;                                   // treatment_distance
    }
}

// ---------------- launcher ----------------

extern "C" void kernel_launch(void* const* d_in, const int* in_sizes, int n_in,
                              void* d_out, int out_size, void* d_ws, size_t ws_size,
                              hipStream_t stream) {
    const float* emb_c = (const float*)d_in[0];
    const float* emb_t = (const float*)d_in[1];
    const int*   user  = (const int*)d_in[2];
    const int*   item  = (const int*)d_in[3];
    const float* label = (const float*)d_in[4];
    const unsigned char* mask = (const unsigned char*)d_in[5];
    const int* src_c = (const int*)d_in[6];
    const int* dst_c = (const int*)d_in[7];
    const int* src_t = (const int*)d_in[8];
    const int* dst_t = (const int*)d_in[9];
    float* out = (float*)d_out;

    // workspace layout
    float* W      = (float*)d_ws;
    float* fc     = W;                  // ND
    float* ftb    = fc  + (size_t)ND;   // ND
    float* hA     = ftb + (size_t)ND;   // ND
    float* hB     = hA  + (size_t)ND;   // ND
    float* norm_c = hB  + (size_t)ND;   // N_NODES
    float* norm_t = norm_c + N_NODES;   // N_NODES
    float* sc     = norm_t + N_NODES;   // NPAIR
    float* st     = sc + NPAIR;         // NPAIR
    float* red    = st + NPAIR;         // 16

    const int T = 256;
    int gND   = (ND + T - 1) / T;
    int gNN   = (N_NODES + T - 1) / T;
    int gE    = (N_EDGES + T - 1) / T;
    int gE16  = (N_EDGES * 16 + T - 1) / T;
    int gP    = (NPAIR + T - 1) / T;
    int gWMMA = (NPAIR / 16) * 32 / T;  // 1280 waves -> 160 blocks

    // degrees -> norms
    k_zero<<<gNN, T, 0, stream>>>(norm_c, N_NODES);
    k_zero<<<gNN, T, 0, stream>>>(norm_t, N_NODES);
    k_zero<<<1, 32, 0, stream>>>(red, 16);
    k_deg<<<gE, T, 0, stream>>>(norm_c, dst_c, N_EDGES);
    k_deg<<<gE, T, 0, stream>>>(norm_t, dst_t, N_EDGES);
    k_norm<<<gNN, T, 0, stream>>>(norm_c, N_NODES);
    k_norm<<<gNN, T, 0, stream>>>(norm_t, N_NODES);

    // LightGCN propagation for both graphs (sequential; shared ping-pong temps)
    const float* embs[2]  = {emb_c, emb_t};
    const int*   srcs[2]  = {src_c, src_t};
    const int*   dsts[2]  = {dst_c, dst_t};
    float*       accs[2]  = {fc, ftb};
    float*       norms[2] = {norm_c, norm_t};

    for (int g = 0; g < 2; ++g) {
        k_copy<<<gND, T, 0, stream>>>(accs[g], embs[g], ND);
        const float* hin = embs[g];
        float* bufS = hA;
        float* bufT = hB;
        for (int l = 0; l < N_LAYERS; ++l) {
            k_scale<<<gND, T, 0, stream>>>(bufS, hin, norms[g], ND);
            k_zero<<<gND, T, 0, stream>>>(bufT, ND);
            k_scatter<<<gE16, T, 0, stream>>>(bufT, bufS, srcs[g], dsts[g], N_EDGES);
            k_scale_add<<<gND, T, 0, stream>>>(accs[g], bufT, norms[g], ND);
            hin = bufT;
            float* tmp = bufS; bufS = bufT; bufT = tmp;
        }
        k_quarter<<<gND, T, 0, stream>>>(accs[g], ND);
    }

    // WMMA dot-product scores
    k_score_wmma<<<gWMMA, T, 0, stream>>>(fc, ftb, user, item, sc, st);

    // reductions + finalize
    k_reduce_loss<<<gP, T, 0, stream>>>(sc, st, label, mask, red);
    k_reduce_disc<<<gP, T, 0, stream>>>(fc, ftb, user, item, red);
    k_finalize<<<1, 32, 0, stream>>>(red, out);
}